// RelPosMulSelfAttention_28174985462172
// MI455X (gfx1250) — compile-verified
//
#include <hip/hip_runtime.h>
#include <hip/hip_bf16.h>
#include <cstdint>

// MI455X / gfx1250: wave32, WMMA (no MFMA). fp32 reference -> use
// V_WMMA_F32_16X16X4_F32 everywhere to keep exact fp32 GEMM numerics
// while staying on the matrix pipe. Tile staging uses the CDNA5 async
// global->LDS path (GLOBAL_LOAD_ASYNC_TO_LDS_B128, ASYNCcnt).

typedef float v2f __attribute__((ext_vector_type(2)));
typedef float v8f __attribute__((ext_vector_type(8)));

#define N_HEADS 16
#define HEAD_D  64
#define S_LEN   1024
#define K_LEN   2048
#define B_SZ    4
#define H_DIM   1024
#define QS      1028   // padded LDS row stride (floats) to avoid 16-way bank conflicts

__device__ __forceinline__ v8f wmma4(v2f a, v2f b, v8f c) {
  // 8 args: (neg_a, A, neg_b, B, c_mod, C, reuse_a, reuse_b)
  return __builtin_amdgcn_wmma_f32_16x16x4_f32(false, a, false, b, (short)0, c,
                                               false, false);
}

// Async global->LDS 16B copy (per lane). LDS byte address = low 32 bits of the
// generic shared-memory pointer.
__device__ __forceinline__ void async_b128(const float* __restrict__ gbase,
                                           unsigned goff_bytes,
                                           const float* ldsp) {
  unsigned ldsa = (unsigned)(uintptr_t)ldsp;
  asm volatile("global_load_async_to_lds_b128 %0, %1, %2"
               :: "v"(ldsa), "v"(goff_bytes), "s"(gbase)
               : "memory");
}

__device__ __forceinline__ void async_fence_barrier() {
  asm volatile("s_wait_asynccnt 0x0" ::: "memory");
  __syncthreads();
}

// ---------------------------------------------------------------------------
// kv_in = concat(token_embedding, mems[0][:, :1024, :]) along seq dim
// ---------------------------------------------------------------------------
__global__ void concat_kv_kernel(const float* __restrict__ tok,
                                 const float* __restrict__ mems,
                                 float* __restrict__ kv) {
  const int64_t n4 = (int64_t)B_SZ * K_LEN * H_DIM / 4;
  for (int64_t t = (int64_t)blockIdx.x * blockDim.x + threadIdx.x; t < n4;
       t += (int64_t)gridDim.x * blockDim.x) {
    int64_t e   = t << 2;
    int     row = (int)(e >> 10);
    int     col = (int)(e & 1023);
    int     b   = row >> 11;
    int     j   = row & 2047;
    float4  val;
    if (j < S_LEN)
      val = *(const float4*)(tok + ((int64_t)(b * S_LEN + j) << 10) + col);
    else
      val = *(const float4*)(mems + ((int64_t)(b * K_LEN + (j - S_LEN)) << 10) + col);
    *(float4*)(kv + e) = val;
  }
}

// ---------------------------------------------------------------------------
// C[M x 1024] = A[M x 1024] @ W^T + bias    (W is 1024x1024, row-major)
// grid = (M/16, 2); block = 256 (8 waves); wave w owns 64 cols (4 x 16-wide
// tiles) so every A fragment is reused by 4 WMMAs.
// ---------------------------------------------------------------------------
__global__ void gemm_bias_kernel(const float* __restrict__ A,
                                 const float* __restrict__ W,
                                 const float* __restrict__ bias,
                                 float* __restrict__ C, int M) {
  (void)M;
  extern __shared__ float sA[];  // 16 * QS floats
  const int t  = threadIdx.x;
  const int m0 = blockIdx.x * 16;
  const int wv = t >> 5, lane = t & 31, lm = lane & 15, lh = lane >> 4;

  // Async-stage 16 rows of A into LDS (no VGPR round-trip)
  for (int it = 0; it < 16; ++it) {
    int idx = it * 256 + t;        // float4 index 0..4095
    int row = idx >> 8;            // 0..15
    int c4  = (idx & 255) << 2;    // 0..1020
    async_b128(A, (unsigned)(((m0 + row) << 12) + (c4 << 2)), &sA[row * QS + c4]);
  }
  async_fence_barrier();

  const int    nb = (blockIdx.y * 8 + wv) * 64;
  const float* w0 = W + ((int64_t)(nb + lm) << 10);
  const float* w1 = w0 + (16 << 10);
  const float* w2 = w1 + (16 << 10);
  const float* w3 = w2 + (16 << 10);
  v8f acc0 = {}, acc1 = {}, acc2 = {}, acc3 = {};
#pragma unroll 2
  for (int ks = 0; ks < 256; ++ks) {
    int    koff = 4 * ks + 2 * lh;
    float2 av   = *(const float2*)&sA[lm * QS + koff];
    v2f a; a.x = av.x; a.y = av.y;
    float2 b0v = *(const float2*)(w0 + koff);
    float2 b1v = *(const float2*)(w1 + koff);
    float2 b2v = *(const float2*)(w2 + koff);
    float2 b3v = *(const float2*)(w3 + koff);
    v2f b0, b1, b2, b3;
    b0.x = b0v.x; b0.y = b0v.y; b1.x = b1v.x; b1.y = b1v.y;
    b2.x = b2v.x; b2.y = b2v.y; b3.x = b3v.x; b3.y = b3v.y;
    acc0 = wmma4(a, b0, acc0);
    acc1 = wmma4(a, b1, acc1);
    acc2 = wmma4(a, b2, acc2);
    acc3 = wmma4(a, b3, acc3);
  }
  const float bc0 = bias[nb + lm];
  const float bc1 = bias[nb + 16 + lm];
  const float bc2 = bias[nb + 32 + lm];
  const float bc3 = bias[nb + 48 + lm];
#pragma unroll
  for (int r = 0; r < 8; ++r) {
    int64_t ro = ((int64_t)(m0 + r + 8 * lh) << 10) + nb + lm;
    C[ro]      = acc0[r] + bc0;
    C[ro + 16] = acc1[r] + bc1;
    C[ro + 32] = acc2[r] + bc2;
    C[ro + 48] = acc3[r] + bc3;
  }
}

// ---------------------------------------------------------------------------
// One 16x16 score tile for head n at (i0, j0):
//   s(ii,jj) = ((q+u)_i . k_j  +  relshift[(q+v)_i . kr_{j+1023-i}]) / 8 - big
// AC via WMMA; BD via a 16x32 band-GEMM + diagonal gather through LDS.
// ---------------------------------------------------------------------------
__device__ __forceinline__ void score_tile(
    const float* __restrict__ Kb, const float* __restrict__ Rb,
    const float* __restrict__ mask, const float* sQ, const float* sU,
    const float* sV, float* sPw, int b, int i0, int j0, int n, int lm, int lh,
    float s[8]) {
  const int hd = n * 64;

  // AC = (q+u)[16x64] x k^T[64x16]
  v8f ac = {};
  {
    const float* krow = Kb + ((int64_t)((j0 + lm) * 4 + b) << 10) + hd;
#pragma unroll 4
    for (int ks = 0; ks < 16; ++ks) {
      int    koff = 4 * ks + 2 * lh;
      float2 qv   = *(const float2*)&sQ[lm * QS + hd + koff];
      float2 uv   = *(const float2*)&sU[hd + koff];
      float2 kv   = *(const float2*)(krow + koff);
      v2f a, bb;
      a.x = qv.x + uv.x; a.y = qv.y + uv.y;
      bb.x = kv.x; bb.y = kv.y;
      ac = wmma4(a, bb, ac);
    }
  }

  // P = (q+v)[16x64] x kr_band^T[64x32], band base = j0+1023-i0-15
  const int base = j0 + (S_LEN - 1) - i0 - 15;
#pragma unroll
  for (int tcol = 0; tcol < 2; ++tcol) {
    int rr = base + tcol * 16 + lm;
    rr     = rr < 0 ? 0 : (rr > K_LEN - 1 ? K_LEN - 1 : rr);
    const float* rrow = Rb + ((int64_t)rr << 10) + hd;
    v8f accp = {};
#pragma unroll 4
    for (int ks = 0; ks < 16; ++ks) {
      int    koff = 4 * ks + 2 * lh;
      float2 qv   = *(const float2*)&sQ[lm * QS + hd + koff];
      float2 vv   = *(const float2*)&sV[hd + koff];
      float2 rv   = *(const float2*)(rrow + koff);
      v2f a, bb;
      a.x = qv.x + vv.x; a.y = qv.y + vv.y;
      bb.x = rv.x; bb.y = rv.y;
      accp = wmma4(a, bb, accp);
    }
#pragma unroll
    for (int r = 0; r < 8; ++r)
      sPw[(r + 8 * lh) * 32 + tcol * 16 + lm] = accp[r];
  }
  asm volatile("s_wait_dscnt 0" ::: "memory");

#pragma unroll
  for (int r = 0; r < 8; ++r) {
    int   M   = r + 8 * lh;
    int   i   = i0 + M, j = j0 + lm;
    int   src = j + (S_LEN - 1) - i;
    float bd  = (src < K_LEN) ? sPw[M * 32 + (15 + lm - M)] : 0.0f;
    float sv  = (ac[r] + bd) * 0.125f;
    // Faithful "big" subtraction through the (B,16,S,K) reinterpretation.
    unsigned L  = (unsigned)n + 16u * ((unsigned)b + 4u * ((unsigned)j + 2048u * (unsigned)i));
    unsigned k2 = L & 2047u;
    unsigned b2 = L >> 25;
    float    mv = (k2 < 1024u) ? 1.0f : mask[b2 * 1024u + (k2 - 1024u)];
    s[r]        = sv - (1.0f - mv) * 1e9f;
  }
}

// ---------------------------------------------------------------------------
// Shared staging for the attention kernels: q rows + u + v into LDS (async).
// ---------------------------------------------------------------------------
__device__ __forceinline__ void stage_q_uv(const float* __restrict__ Qb,
                                           const float* __restrict__ Ub,
                                           const float* __restrict__ Vhb,
                                           float* sQ, float* sU, float* sV,
                                           int i0, int b, int t) {
  for (int it = 0; it < 16; ++it) {
    int idx = it * 256 + t;
    int row = idx >> 8;
    int c4  = (idx & 255) << 2;
    async_b128(Qb, (unsigned)((((i0 + row) * 4 + b) << 12) + (c4 << 2)),
               &sQ[row * QS + c4]);
  }
  async_b128(Ub, (unsigned)(t << 4), &sU[t << 2]);
  async_b128(Vhb, (unsigned)(t << 4), &sV[t << 2]);
  async_fence_barrier();
}

// ---------------------------------------------------------------------------
// Pass 1: Z[b,i,j] = sum over heads of exp(s). grid=(64,4), block=256.
// Each of 8 waves handles 2 heads; exp-sums merged with LDS ds_add_f32.
// ---------------------------------------------------------------------------
__global__ void attn_z_kernel(const float* __restrict__ Qb,
                              const float* __restrict__ Kb,
                              const float* __restrict__ Rb,
                              const float* __restrict__ Ub,
                              const float* __restrict__ Vhb,
                              const float* __restrict__ mask,
                              float* __restrict__ Zb) {
  extern __shared__ float smem[];
  float* sQ  = smem;                // 16*QS
  float* sU  = sQ + 16 * QS;        // 1024
  float* sV  = sU + 1024;           // 1024
  float* sP  = sV + 1024;           // 8*16*32
  float* sZt = sP + 8 * 16 * 32;    // 256

  const int t  = threadIdx.x;
  const int wv = t >> 5, lane = t & 31, lm = lane & 15, lh = lane >> 4;
  const int i0 = blockIdx.x * 16;
  const int b  = blockIdx.y;

  stage_q_uv(Qb, Ub, Vhb, sQ, sU, sV, i0, b, t);

  float* sPw = sP + wv * (16 * 32);
  for (int jt = 0; jt < 128; ++jt) {
    const int j0 = jt * 16;
    sZt[t]       = 0.0f;
    __syncthreads();
#pragma unroll
    for (int hh = 0; hh < 2; ++hh) {
      int   n = wv * 2 + hh;
      float s[8];
      score_tile(Kb, Rb, mask, sQ, sU, sV, sPw, b, i0, j0, n, lm, lh, s);
#pragma unroll
      for (int r = 0; r < 8; ++r) {
        int M = r + 8 * lh;
        atomicAdd(&sZt[M * 16 + lm], __expf(s[r]));  // ds_add_f32
      }
    }
    __syncthreads();
    {
      int M = t >> 4, N = t & 15;
      Zb[((int64_t)(b * S_LEN + i0 + M) << 11) + j0 + N] = sZt[t];
    }
    __syncthreads();
  }
}

// ---------------------------------------------------------------------------
// Pass 2: out[i,b,n,:] = sum_j exp(s)/Z * v.  grid=(64,2,4), block=256,
// wave w owns head hg*8+w; P.V accumulated with WMMA (K=16 per j-tile).
// ---------------------------------------------------------------------------
__global__ void attn_out_kernel(const float* __restrict__ Qb,
                                const float* __restrict__ Kb,
                                const float* __restrict__ Rb,
                                const float* __restrict__ Vb,
                                const float* __restrict__ Ub,
                                const float* __restrict__ Vhb,
                                const float* __restrict__ mask,
                                const float* __restrict__ Zb,
                                float* __restrict__ AO) {
  extern __shared__ float smem[];
  float* sQ  = smem;                // 16*QS
  float* sU  = sQ + 16 * QS;        // 1024
  float* sV  = sU + 1024;           // 1024
  float* sP  = sV + 1024;           // 8*16*32
  float* sPm = sP + 8 * 16 * 32;    // 8*16*16

  const int t  = threadIdx.x;
  const int wv = t >> 5, lane = t & 31, lm = lane & 15, lh = lane >> 4;
  const int i0 = blockIdx.x * 16;
  const int hg = blockIdx.y;
  const int b  = blockIdx.z;
  const int n  = hg * 8 + wv;

  stage_q_uv(Qb, Ub, Vhb, sQ, sU, sV, i0, b, t);

  float* sPw  = sP + wv * (16 * 32);
  float* sPmw = sPm + wv * 256;
  v8f    acc0 = {}, acc1 = {}, acc2 = {}, acc3 = {};

  for (int jt = 0; jt < 128; ++jt) {
    const int j0 = jt * 16;
    float     s[8];
    score_tile(Kb, Rb, mask, sQ, sU, sV, sPw, b, i0, j0, n, lm, lh, s);
#pragma unroll
    for (int r = 0; r < 8; ++r) {
      int   M = r + 8 * lh;
      float z = Zb[((int64_t)(b * S_LEN + i0 + M) << 11) + j0 + lm];
      sPmw[M * 16 + lm] = __expf(s[r]) / z;
    }
    asm volatile("s_wait_dscnt 0" ::: "memory");
    // acc += P[16x16] x V[16x64]  (4 K-steps x 4 d-subtiles)
#pragma unroll
    for (int ks = 0; ks < 4; ++ks) {
      int kk = 4 * ks + 2 * lh;  // this lane's two B-rows (j-local)
      v2f a;
      a.x = sPmw[lm * 16 + kk];
      a.y = sPmw[lm * 16 + kk + 1];
      const float* v0 = Vb + ((int64_t)((j0 + kk) * 4 + b) << 10) + n * 64;
      const float* v1 = Vb + ((int64_t)((j0 + kk + 1) * 4 + b) << 10) + n * 64;
      v2f b0, b1, b2, b3;
      b0.x = v0[lm];      b0.y = v1[lm];
      b1.x = v0[16 + lm]; b1.y = v1[16 + lm];
      b2.x = v0[32 + lm]; b2.y = v1[32 + lm];
      b3.x = v0[48 + lm]; b3.y = v1[48 + lm];
      acc0 = wmma4(a, b0, acc0);
      acc1 = wmma4(a, b1, acc1);
      acc2 = wmma4(a, b2, acc2);
      acc3 = wmma4(a, b3, acc3);
    }
    asm volatile("" ::: "memory");
  }

#pragma unroll
  for (int r = 0; r < 8; ++r) {
    int     M      = r + 8 * lh;
    int64_t rowoff = ((int64_t)((i0 + M) * 4 + b) << 10) + n * 64 + lm;
    AO[rowoff]      = acc0[r];
    AO[rowoff + 16] = acc1[r];
    AO[rowoff + 32] = acc2[r];
    AO[rowoff + 48] = acc3[r];
  }
}

// ---------------------------------------------------------------------------
// mems_new: copy mems, overwrite rows [1024,2048) per b with the final output
// ---------------------------------------------------------------------------
__global__ void mems_out_kernel(const float* __restrict__ mems,
                                const float* __restrict__ outf,
                                float* __restrict__ mo) {
  const int64_t n4 = (int64_t)B_SZ * K_LEN * H_DIM / 4;
  for (int64_t t = (int64_t)blockIdx.x * blockDim.x + threadIdx.x; t < n4;
       t += (int64_t)gridDim.x * blockDim.x) {
    int64_t e   = t << 2;
    int     b   = (int)(e >> 21);
    int     rem = (int)(e & ((1 << 21) - 1));
    int     r   = rem >> 10;
    int     c   = rem & 1023;
    float4  val;
    if (r >= S_LEN)
      val = *(const float4*)(outf + ((int64_t)(b * S_LEN + (r - S_LEN)) << 10) + c);
    else
      val = *(const float4*)(mems + e);
    *(float4*)(mo + e) = val;
  }
}

// ---------------------------------------------------------------------------
extern "C" void kernel_launch(void* const* d_in, const int* in_sizes, int n_in,
                              void* d_out, int out_size, void* d_ws,
                              size_t ws_size, hipStream_t stream) {
  (void)in_sizes; (void)n_in; (void)out_size; (void)ws_size;
  const float* tok  = (const float*)d_in[0];
  const float* rel  = (const float*)d_in[1];
  const float* mems = (const float*)d_in[2];
  const float* mask = (const float*)d_in[3];
  const float* Wq = (const float*)d_in[4];  const float* bq = (const float*)d_in[5];
  const float* Wk = (const float*)d_in[6];  const float* bk = (const float*)d_in[7];
  const float* Wv = (const float*)d_in[8];  const float* bv = (const float*)d_in[9];
  const float* Wr = (const float*)d_in[10]; const float* br = (const float*)d_in[11];
  const float* Wo = (const float*)d_in[12]; const float* bo = (const float*)d_in[13];
  const float* u  = (const float*)d_in[14];
  const float* vh = (const float*)d_in[15];
  float* out = (float*)d_out;

  // Workspace layout (floats); total 44,040,192 floats = 176 MB.
  float* ws = (float*)d_ws;
  float* KV = ws;                                   // 8192x1024
  float* Qb = KV + (int64_t)8192 * 1024;            // 4096x1024
  float* Kb = Qb + (int64_t)4096 * 1024;            // 8192x1024
  float* Vb = Kb + (int64_t)8192 * 1024;            // 8192x1024
  float* Rb = Vb + (int64_t)8192 * 1024;            // 2048x1024
  float* Zb = Rb + (int64_t)2048 * 1024;            // 4x1024x2048
  float* AO = Zb + (int64_t)4 * 1024 * 2048;        // 4096x1024

  const size_t gemm_lds = (size_t)16 * QS * sizeof(float);

  concat_kv_kernel<<<2048, 256, 0, stream>>>(tok, mems, KV);

  gemm_bias_kernel<<<dim3(256, 2), 256, gemm_lds, stream>>>(tok, Wq, bq, Qb, 4096);
  gemm_bias_kernel<<<dim3(512, 2), 256, gemm_lds, stream>>>(KV, Wk, bk, Kb, 8192);
  gemm_bias_kernel<<<dim3(512, 2), 256, gemm_lds, stream>>>(KV, Wv, bv, Vb, 8192);
  gemm_bias_kernel<<<dim3(128, 2), 256, gemm_lds, stream>>>(rel, Wr, br, Rb, 2048);

  const size_t z_lds = (size_t)(16 * QS + 1024 + 1024 + 8 * 16 * 32 + 256) * sizeof(float);
  attn_z_kernel<<<dim3(64, 4), 256, z_lds, stream>>>(Qb, Kb, Rb, u, vh, mask, Zb);

  const size_t o_lds = (size_t)(16 * QS + 1024 + 1024 + 8 * 16 * 32 + 8 * 16 * 16) * sizeof(float);
  attn_out_kernel<<<dim3(64, 2, 4), 256, o_lds, stream>>>(Qb, Kb, Rb, Vb, u, vh,
                                                          mask, Zb, AO);

  gemm_bias_kernel<<<dim3(256, 2), 256, gemm_lds, stream>>>(AO, Wo, bo, out, 4096);

  mems_out_kernel<<<2048, 256, 0, stream>>>(mems, out,
                                            out + (int64_t)4 * 1024 * 1024);
}